// Proposal_Target_Layer_60404420051302
// MI455X (gfx1250) — compile-verified
//
#include <hip/hip_runtime.h>
#include <hip/hip_bf16.h>

// ---------------------------------------------------------------------------
// Proposal Target Layer for MI455X (gfx1250, wave32).
// VALU/issue-bound (2 x 25.6M IoU evals) over an HBM floor of ~0.35 us;
// WMMA not applicable (max/min/compare/divide, no inner products).
// CDNA5-specific paths:
//   * global_load_async_to_lds_b32/b128 + s_wait_asynccnt (staging)
//   * global_atomic_max_u64 on packed (iou_bits<<32 | ~idx) keys
//     (deterministic column argmax, exact first-index tie-break)
//   * wave32 ballot prefix scans (__builtin_amdgcn_ballot_w32)
// ---------------------------------------------------------------------------

#define G_MAX      128
#define POS_THR    0.7f
#define POS_CAP    64
#define NEG_CAP    192
#define BG_CLS     20.0f   // CLS_NUM - 1

__device__ __forceinline__ void async_gld_b32_to_lds(const float* gptr, float* lptr) {
    // Low 32 bits of the generic (LDS-aperture) address are the allocation-
    // relative LDS byte offset (ISA 10.2).
    unsigned lds_off = (unsigned)(unsigned long long)(uintptr_t)lptr;
    asm volatile("global_load_async_to_lds_b32 %0, %1, off"
                 :: "v"(lds_off), "v"(gptr) : "memory");
}

__device__ __forceinline__ void async_gld_b128_to_lds(const float* gptr, float* lptr) {
    unsigned lds_off = (unsigned)(unsigned long long)(uintptr_t)lptr;
    asm volatile("global_load_async_to_lds_b128 %0, %1, off"
                 :: "v"(lds_off), "v"(gptr) : "memory");
}

__device__ __forceinline__ void wait_asynccnt0() {
    asm volatile("s_wait_asynccnt 0" ::: "memory");
}

// Stage G*5 dwords of the GT table into LDS (vectorized b128 + b32 tail).
__device__ __forceinline__ void stage_gts(const float* __restrict__ gts, float* sraw,
                                          int G, int t, int nthr) {
    int tot_dw = G * 5;
    int nvec = tot_dw >> 2;
    for (int k = t; k < nvec; k += nthr)
        async_gld_b128_to_lds(gts + 4 * k, &sraw[4 * k]);
    for (int k = (nvec << 2) + t; k < tot_dw; k += nthr)
        async_gld_b32_to_lds(gts + k, &sraw[k]);
}

// Repack raw (stride-5) GT records into 32B-padded {x1,y1,x2,y2,area,cls,_,_}.
__device__ __forceinline__ void repack_gts(const float* sraw, float* sgt8, int G, int t) {
    if (t < G) {
        float x1 = sraw[t * 5 + 0], y1 = sraw[t * 5 + 1];
        float x2 = sraw[t * 5 + 2], y2 = sraw[t * 5 + 3];
        float c  = sraw[t * 5 + 4];
        sgt8[t * 8 + 0] = x1; sgt8[t * 8 + 1] = y1;
        sgt8[t * 8 + 2] = x2; sgt8[t * 8 + 3] = y2;
        sgt8[t * 8 + 4] = (x2 - x1) * (y2 - y1);
        sgt8[t * 8 + 5] = c;
    }
}

__device__ __forceinline__ float iou_box(float px1, float py1, float px2, float py2,
                                         float area_p,
                                         float gx1, float gy1, float gx2, float gy2,
                                         float area_g) {
    float xx1 = fmaxf(px1, gx1), yy1 = fmaxf(py1, gy1);
    float xx2 = fminf(px2, gx2), yy2 = fminf(py2, gy2);
    float iw = fmaxf(xx2 - xx1, 0.0f), ih = fmaxf(yy2 - yy1, 0.0f);
    float inter = iw * ih;
    return inter / (area_p + area_g - inter);   // IEEE div, matches jnp
}

// ---------------------------------------------------------------------------
// K0: prefill pos_keep/neg_keep with -1, zero column-argmax keys.
// ---------------------------------------------------------------------------
__global__ void k_init(float* __restrict__ out, unsigned long long* __restrict__ gcol, int G) {
    int t = threadIdx.x;
    if (t < POS_CAP + NEG_CAP) out[t] = -1.0f;
    if (t < G) gcol[t] = 0ULL;
}

// ---------------------------------------------------------------------------
// K1: row pass. Anchor argmax + direct threshold masks (anchor_max never
// leaves the kernel). GT table async-staged, repacked to 32B records so the
// hot loop is ds_load_b128 + ds_load_b32 per GT (wave-uniform -> broadcast).
// ---------------------------------------------------------------------------
__global__ void k_row(const float4* __restrict__ prop, const float* __restrict__ gts,
                      int N, int G,
                      int* __restrict__ aidx,
                      unsigned char* __restrict__ posm, unsigned char* __restrict__ negm) {
    __shared__ __align__(16) float sraw[G_MAX * 5];
    __shared__ __align__(16) float sgt8[G_MAX * 8];

    int t = threadIdx.x;
    stage_gts(gts, sraw, G, t, blockDim.x);
    wait_asynccnt0();
    __syncthreads();
    repack_gts(sraw, sgt8, G, t);
    __syncthreads();

    int i = blockIdx.x * blockDim.x + t;
    if (i >= N) return;

    float4 p = prop[i];
    float area_p = (p.z - p.x) * (p.w - p.y);
    float best = -1.0f;
    int   bi   = 0;
    for (int g = 0; g < G; ++g) {
        const float4 gb = *reinterpret_cast<const float4*>(&sgt8[g * 8]);
        float area_g = sgt8[g * 8 + 4];
        float iou = iou_box(p.x, p.y, p.z, p.w, area_p,
                            gb.x, gb.y, gb.z, gb.w, area_g);
        if (iou > best) { best = iou; bi = g; }   // strict > == first-index ties
    }
    aidx[i] = bi;
    posm[i] = (best > POS_THR) ? 1 : 0;
    negm[i] = (best < POS_THR) ? 1 : 0;
}

// ---------------------------------------------------------------------------
// K2: column argmax, transposed parallelism. Lane owns one GT box
// (g = (wid&3)*32 + lane); waves 0-3 scan chunk rows [0,128), waves 4-7 scan
// [128,256). Proposal reads are one uniform ds_load_b128 (broadcast); per-lane
// best kept in registers -> zero inner-loop atomics. Candidates merged via LDS
// (strict > keeps lower-index half), then ONE packed-key global atomic_max_u64
// per (block, g).
// ---------------------------------------------------------------------------
__global__ void k_col(const float* __restrict__ prop_flat, const float* __restrict__ gts,
                      int N, int G,
                      unsigned long long* __restrict__ gcol) {
    __shared__ __align__(16) float sraw[G_MAX * 5];
    __shared__ __align__(16) float sprop[256 * 4];
    __shared__ float scv[G_MAX];
    __shared__ int   sci[G_MAX];

    int t = threadIdx.x;
    int base = blockIdx.x * 256;
    int count = N - base; if (count > 256) count = 256;

    stage_gts(gts, sraw, G, t, 256);
    for (int k = t; k < count; k += 256)                      // one float4 per proposal
        async_gld_b128_to_lds(prop_flat + (size_t)(base + k) * 4, &sprop[k * 4]);
    wait_asynccnt0();
    __syncthreads();

    int lane = t & 31, wid = t >> 5;
    int g = ((wid & 3) << 5) | lane;           // 0..127
    bool act = g < G;

    float gx1 = 0.f, gy1 = 0.f, gx2 = 0.f, gy2 = 0.f, area_g = 0.f;
    if (act) {
        gx1 = sraw[g * 5 + 0]; gy1 = sraw[g * 5 + 1];
        gx2 = sraw[g * 5 + 2]; gy2 = sraw[g * 5 + 3];
        area_g = (gx2 - gx1) * (gy2 - gy1);
    }

    int j0 = (wid < 4) ? 0 : 128;
    int j1 = (wid < 4) ? (count < 128 ? count : 128) : count;

    float best = -1.0f;
    int   bi   = 0x7FFFFFFF;
    if (act) {
        for (int j = j0; j < j1; ++j) {
            const float4 pb = *reinterpret_cast<const float4*>(&sprop[j * 4]);
            float area_p = (pb.z - pb.x) * (pb.w - pb.y);
            float iou = iou_box(pb.x, pb.y, pb.z, pb.w, area_p,
                                gx1, gy1, gx2, gy2, area_g);
            if (iou > best) { best = iou; bi = base + j; }  // ascending j -> first idx
        }
    }

    if (act && wid < 4) { scv[g] = best; sci[g] = bi; }     // unique lane per g
    __syncthreads();
    if (act && wid >= 4) {
        // upper half always has larger indices: strict > keeps lower on ties
        if (best > scv[g]) { scv[g] = best; sci[g] = bi; }
    }
    __syncthreads();
    if (t < G && sci[t] != 0x7FFFFFFF) {
        unsigned long long key =
            ((unsigned long long)__float_as_uint(scv[t]) << 32) |
            (unsigned long long)(~(unsigned)sci[t]);        // max ~i == min i
        atomicMax(&gcol[t], key);                           // global_atomic_max_u64
    }
}

// ---------------------------------------------------------------------------
// K3: force best proposal per GT positive / not negative.
// ---------------------------------------------------------------------------
__global__ void k_force(const unsigned long long* __restrict__ gcol, int G,
                        unsigned char* __restrict__ posm, unsigned char* __restrict__ negm) {
    int g = threadIdx.x;
    if (g < G) {
        unsigned i = ~(unsigned)(gcol[g] & 0xFFFFFFFFULL);
        posm[i] = 1;    // idempotent across duplicate g -> same i
        negm[i] = 0;
    }
}

// ---------------------------------------------------------------------------
// K4: per-block (256 items) counts via wave32 ballots.
// ---------------------------------------------------------------------------
__global__ void k_count(const unsigned char* __restrict__ posm,
                        const unsigned char* __restrict__ negm, int N,
                        int* __restrict__ blkp, int* __restrict__ blkn) {
    __shared__ int sp[8], sn[8];
    int t = threadIdx.x;
    int i = blockIdx.x * blockDim.x + t;
    bool p = (i < N) && posm[i];
    bool n = (i < N) && negm[i];
    unsigned bp = __builtin_amdgcn_ballot_w32(p);
    unsigned bn = __builtin_amdgcn_ballot_w32(n);
    int lane = t & 31, wid = t >> 5;
    if (lane == 0) { sp[wid] = __popc(bp); sn[wid] = __popc(bn); }
    __syncthreads();
    if (t == 0) {
        int tp = 0, tn = 0;
        for (int w = 0; w < 8; ++w) { tp += sp[w]; tn += sn[w]; }
        blkp[blockIdx.x] = tp;
        blkn[blockIdx.x] = tn;
    }
}

// ---------------------------------------------------------------------------
// K5: single-block exclusive scan of block counts (nblk <= 1024) + totals.
// ---------------------------------------------------------------------------
__global__ void k_scan(const int* __restrict__ blkp, const int* __restrict__ blkn, int nblk,
                       int* __restrict__ offp, int* __restrict__ offn, int* __restrict__ tot) {
    __shared__ int sp[1024], sn[1024];
    int t = threadIdx.x;
    int vp0 = (t < nblk) ? blkp[t] : 0;
    int vn0 = (t < nblk) ? blkn[t] : 0;
    sp[t] = vp0; sn[t] = vn0;
    __syncthreads();
    for (int d = 1; d < 1024; d <<= 1) {
        int ap = 0, an = 0;
        if (t >= d) { ap = sp[t - d]; an = sn[t - d]; }
        __syncthreads();
        sp[t] += ap; sn[t] += an;
        __syncthreads();
    }
    if (t < nblk) { offp[t] = sp[t] - vp0; offn[t] = sn[t] - vn0; }
    if (t == 1023) {
        int total_pos   = sp[1023];
        int runtime_pos = total_pos < POS_CAP ? total_pos : POS_CAP;
        // floor(rp / 0.25 * 0.75) == 3*rp exactly in fp32 for rp <= 64
        int neg_bound = (int)floorf((float)runtime_pos / 0.25f * 0.75f);
        tot[0] = total_pos;
        tot[1] = runtime_pos;
        tot[2] = neg_bound;
    }
}

// ---------------------------------------------------------------------------
// K6: finalize. Ballot-based ordered ranks, keep lists, cls + reg targets.
// out layout: [0..63] pos_keep, [64..255] neg_keep, [256..256+N) cls,
//             [256+N .. 256+5N) reg (row-major Nx4). All as float.
// ---------------------------------------------------------------------------
__global__ void k_final(const float4* __restrict__ prop, const float* __restrict__ gts,
                        const int* __restrict__ aidx,
                        const unsigned char* __restrict__ posm,
                        const unsigned char* __restrict__ negm,
                        const int* __restrict__ offp, const int* __restrict__ offn,
                        const int* __restrict__ tot,
                        int N, int G, float* __restrict__ out) {
    __shared__ __align__(16) float sraw[G_MAX * 5];
    __shared__ __align__(16) float sgt8[G_MAX * 8];
    __shared__ int wp[8], wn[8];

    int t = threadIdx.x;
    stage_gts(gts, sraw, G, t, blockDim.x);

    int i = blockIdx.x * blockDim.x + t;
    bool p = (i < N) && posm[i];
    bool n = (i < N) && negm[i];
    unsigned bp = __builtin_amdgcn_ballot_w32(p);
    unsigned bn = __builtin_amdgcn_ballot_w32(n);
    int lane = t & 31, wid = t >> 5;
    if (lane == 0) { wp[wid] = __popc(bp); wn[wid] = __popc(bn); }
    wait_asynccnt0();
    __syncthreads();
    repack_gts(sraw, sgt8, G, t);
    __syncthreads();

    if (i >= N) return;

    int pofs = 0, nofs = 0;
    for (int w = 0; w < wid; ++w) { pofs += wp[w]; nofs += wn[w]; }
    unsigned lmask = (1u << lane) - 1u;
    int prank = offp[blockIdx.x] + pofs + __popc(bp & lmask);
    int nrank = offn[blockIdx.x] + nofs + __popc(bn & lmask);

    bool kept = p && (prank < POS_CAP);
    if (kept) out[prank] = (float)i;
    int neg_bound = tot[2];
    if (n && nrank < neg_bound) out[POS_CAP + nrank] = (float)i;

    int gi = aidx[i];
    out[POS_CAP + NEG_CAP + i] = kept ? sgt8[gi * 8 + 5] : BG_CLS;

    float dx = 0.0f, dy = 0.0f, dw = 0.0f, dh = 0.0f;
    if (kept) {
        float4 pb = prop[i];
        const float4 gb = *reinterpret_cast<const float4*>(&sgt8[gi * 8]);
        float aw = pb.z - pb.x, ah = pb.w - pb.y;
        float acx = pb.x + aw * 0.5f, acy = pb.y + ah * 0.5f;
        float gw = gb.z - gb.x, gh = gb.w - gb.y;
        float gcx = gb.x + gw * 0.5f, gcy = gb.y + gh * 0.5f;
        dx = (gcx - acx) / aw;
        dy = (gcy - acy) / ah;
        dw = logf(gw / aw);
        dh = logf(gh / ah);
    }
    float* r = out + (size_t)(POS_CAP + NEG_CAP) + (size_t)N + (size_t)4 * i;
    r[0] = dx; r[1] = dy; r[2] = dw; r[3] = dh;
}

// ---------------------------------------------------------------------------
extern "C" void kernel_launch(void* const* d_in, const int* in_sizes, int n_in,
                              void* d_out, int out_size, void* d_ws, size_t ws_size,
                              hipStream_t stream) {
    const float* prop = (const float*)d_in[0];   // [N,4]
    const float* gts  = (const float*)d_in[1];   // [G,5]
    int N = in_sizes[0] / 4;
    int G = in_sizes[1] / 5;
    float* out = (float*)d_out;

    char* ws = (char*)d_ws;
    size_t o = 0;
    int*   aidx = (int*)(ws + o);                   o += (size_t)N * 4;
    unsigned char* posm = (unsigned char*)(ws + o); o += (size_t)N;
    unsigned char* negm = (unsigned char*)(ws + o); o += (size_t)N;
    o = (o + 15) & ~(size_t)15;
    unsigned long long* gcol = (unsigned long long*)(ws + o); o += (size_t)G * 8;
    int nblk = (N + 255) / 256;                  // 782 for N=200000 (<=1024)
    int* blkp = (int*)(ws + o); o += (size_t)nblk * 4;
    int* blkn = (int*)(ws + o); o += (size_t)nblk * 4;
    int* offp = (int*)(ws + o); o += (size_t)nblk * 4;
    int* offn = (int*)(ws + o); o += (size_t)nblk * 4;
    int* tot  = (int*)(ws + o); o += 16;

    k_init <<<1,    256,  0, stream>>>(out, gcol, G);
    k_row  <<<nblk, 256,  0, stream>>>((const float4*)prop, gts, N, G, aidx, posm, negm);
    k_col  <<<nblk, 256,  0, stream>>>(prop, gts, N, G, gcol);
    k_force<<<1,    G,    0, stream>>>(gcol, G, posm, negm);
    k_count<<<nblk, 256,  0, stream>>>(posm, negm, N, blkp, blkn);
    k_scan <<<1,    1024, 0, stream>>>(blkp, blkn, nblk, offp, offn, tot);
    k_final<<<nblk, 256,  0, stream>>>((const float4*)prop, gts, aidx, posm, negm,
                                       offp, offn, tot, N, G, out);
}